// ParallelSearch_58213986730357
// MI455X (gfx1250) — compile-verified
//
#include <hip/hip_runtime.h>
#include <hip/hip_bf16.h>

typedef __attribute__((ext_vector_type(2))) float v2f;
typedef __attribute__((ext_vector_type(8))) float v8f;

#define J_DIM 256
#define C_DIM 1000
#define TL    32
#define NB    5
#define KFLAT (TL * J_DIM)   // 8192 flattened (t,j)
#define GP    1008           // padded C dimension: 63 tiles * 16

// Workspace layout (float offsets). Total = 59648 floats = ~233 KB.
#define OFF_SOSW  0                      // 256
#define OFF_D0    256                    // 1000
#define OFF_NBEST 1280                   // 8 ints (uses 5)
#define OFF_P     1536                   // NB*KFLAT = 40960
#define OFF_SP    (OFF_P + NB * KFLAT)   // 42496 (+16)
#define OFF_SL    (OFF_SP + 16)          // 42512 (+GP)
#define OFF_G     (OFF_SL + GP)          // 43520 (+16*GP = 16128)

static __device__ inline float blockReduceSum256(float v) {
    __shared__ float s[256];
    s[threadIdx.x] = v;
    __syncthreads();
    for (int off = 128; off > 0; off >>= 1) {
        if ((int)threadIdx.x < off) s[threadIdx.x] += s[threadIdx.x + off];
        __syncthreads();
    }
    return s[0];
}

// ---- 1. sosW[t] = sum_j sos[j] * W[j][t] ------------------------------------
__global__ void sosw_kernel(const float* __restrict__ W,
                            const float* __restrict__ sos,
                            float* __restrict__ ws) {
    const int t = threadIdx.x;
    float acc = 0.f;
    for (int j = 0; j < J_DIM; ++j)
        acc = fmaf(sos[j], W[j * J_DIM + t], acc);
    ws[OFF_SOSW + t] = acc;
}

// ---- 2. d0[c] = ||sosW - label_seqs[c,0,:]||^2 ------------------------------
__global__ void d0_kernel(const float* __restrict__ L, float* __restrict__ ws) {
    const int c = blockIdx.x, j = threadIdx.x;
    const float diff = ws[OFF_SOSW + j] - L[(size_t)c * KFLAT + j];
    const float s = blockReduceSum256(diff * diff);
    if (j == 0) ws[OFF_D0 + c] = s;
}

// ---- 3. top-5 smallest d0 (first-index tie-break, matches lax.top_k) --------
__global__ void top5_kernel(float* __restrict__ ws) {
    int* nb = (int*)(ws + OFF_NBEST);
    const float* d0 = ws + OFF_D0;
    int chosen[NB];
    for (int k = 0; k < NB; ++k) {
        float best = 3.4e38f;
        int bi = 0;
        for (int c = 0; c < C_DIM; ++c) {
            bool skip = false;
            for (int p = 0; p < k; ++p) skip |= (chosen[p] == c);
            if (!skip && d0[c] < best) { best = d0[c]; bi = c; }
        }
        chosen[k] = bi;
        nb[k] = bi;
    }
}

// ---- 4. P[n] = label_seqs[n_best[n]] @ W  (32x256 @ 256x256, f32 WMMA) ------
// One wave32 per block. grid = (Ntiles=16, Mtiles=2, beams=5).
__global__ void beam_gemm_kernel(const float* __restrict__ L,
                                 const float* __restrict__ W,
                                 float* __restrict__ ws) {
    const int lane = threadIdx.x;
    const int half = lane >> 4;        // 0: lanes 0-15, 1: lanes 16-31
    const int l    = lane & 15;
    const int n0   = blockIdx.x * 16;  // output col (jout)
    const int m0   = blockIdx.y * 16;  // output row (t)
    const int beam = blockIdx.z;
    const int sel  = ((const int*)(ws + OFF_NBEST))[beam];

    const float* A = L + (size_t)sel * KFLAT;  // 32x256 row-major
    const int arow = m0 + l;
    const int bcol = n0 + l;

    v8f acc = {};
    for (int k = 0; k < J_DIM; k += 4) {
        const int kk = k + 2 * half;   // ISA 16x4 A / 4x16 B lane layout
        v2f a, b;
        a.x = A[arow * J_DIM + kk];
        a.y = A[arow * J_DIM + kk + 1];
        b.x = W[kk * J_DIM + bcol];
        b.y = W[(kk + 1) * J_DIM + bcol];
        acc = __builtin_amdgcn_wmma_f32_16x16x4_f32(
            false, a, false, b, (short)0, acc, false, false);
    }

    float* P = ws + OFF_P + (size_t)beam * KFLAT;
    for (int r = 0; r < 8; ++r)
        P[(m0 + r + 8 * half) * J_DIM + (n0 + l)] = acc[r];  // D: VGPR r -> row r/ r+8
}

// ---- 5. row sum-of-squares (used for both SL over L and SP over P) ----------
__global__ void rowsumsq_kernel(const float* __restrict__ src, int rowlen,
                                float* __restrict__ out) {
    const int row = blockIdx.x;
    const float* p = src + (size_t)row * rowlen;
    float acc = 0.f;
    for (int i = threadIdx.x; i < rowlen; i += blockDim.x) {
        const float v = p[i];
        acc = fmaf(v, v, acc);
    }
    const float s = blockReduceSum256(acc);
    if (threadIdx.x == 0) out[row] = s;
}

// ---- 6. G[n,c] = <P[n], label_seqs[c]> over 8192 elems (16x1008 GEMM) -------
// One wave32 per block, grid = 63 column tiles. Rows 5..15 and cols >=1000 are
// padding: clamp indices (no EXEC divergence; garbage rows ignored later).
__global__ void score_gemm_kernel(const float* __restrict__ L,
                                  float* __restrict__ ws) {
    const int lane = threadIdx.x;
    const int half = lane >> 4;
    const int l    = lane & 15;
    const int c0   = blockIdx.x * 16;

    const int arow = (l < NB) ? l : (NB - 1);
    const int c    = c0 + l;
    const int cc   = (c < C_DIM) ? c : (C_DIM - 1);

    const float* A = ws + OFF_P + (size_t)arow * KFLAT;
    const float* B = L + (size_t)cc * KFLAT;

    v8f acc0 = {}, acc1 = {};
    for (int k = 0; k < KFLAT; k += 8) {   // two accumulators break WMMA RAW chain
        {
            const int kk = k + 2 * half;
            v2f a, b;
            a.x = A[kk];     a.y = A[kk + 1];
            b.x = B[kk];     b.y = B[kk + 1];
            acc0 = __builtin_amdgcn_wmma_f32_16x16x4_f32(
                false, a, false, b, (short)0, acc0, false, false);
        }
        {
            const int kk = k + 4 + 2 * half;
            v2f a, b;
            a.x = A[kk];     a.y = A[kk + 1];
            b.x = B[kk];     b.y = B[kk + 1];
            acc1 = __builtin_amdgcn_wmma_f32_16x16x4_f32(
                false, a, false, b, (short)0, acc1, false, false);
        }
    }
    const v8f acc = acc0 + acc1;

    float* G = ws + OFF_G;
    for (int r = 0; r < 8; ++r)
        G[(r + 8 * half) * GP + c0 + l] = acc[r];
}

// ---- 7. score[n,c] = SP[n] + SL[c] - 2 G[n,c]; nested argmins; broadcast ----
__global__ void final_kernel(const float* __restrict__ ws, int* __restrict__ out) {
    __shared__ float sv[NB][256];
    __shared__ int   si[NB][256];
    const int tid = threadIdx.x;
    const float* SP = ws + OFF_SP;
    const float* SL = ws + OFF_SL;
    const float* G  = ws + OFF_G;

    float bv[NB];
    int   bi[NB];
    for (int n = 0; n < NB; ++n) { bv[n] = 3.4e38f; bi[n] = 0; }

    for (int c = tid; c < C_DIM; c += 256) {
        const float sl = SL[c];
        for (int n = 0; n < NB; ++n) {
            const float s = SP[n] + sl - 2.f * G[n * GP + c];
            if (s < bv[n]) { bv[n] = s; bi[n] = c; }   // strict < keeps first c
        }
    }
    for (int n = 0; n < NB; ++n) { sv[n][tid] = bv[n]; si[n][tid] = bi[n]; }
    __syncthreads();
    for (int off = 128; off > 0; off >>= 1) {
        if (tid < off) {
            for (int n = 0; n < NB; ++n) {
                const float v2 = sv[n][tid + off];
                const int   i2 = si[n][tid + off];
                if (v2 < sv[n][tid] || (v2 == sv[n][tid] && i2 < si[n][tid])) {
                    sv[n][tid] = v2;
                    si[n][tid] = i2;
                }
            }
        }
        __syncthreads();
    }
    if (tid == 0) {
        float bestV = sv[0][0];
        int bestN = 0;
        for (int n = 1; n < NB; ++n)
            if (sv[n][0] < bestV) { bestV = sv[n][0]; bestN = n; }  // first-min
        const int label = si[bestN][0];
        for (int b = 0; b < 32; ++b) out[b] = label;  // identical for all batches
    }
}

extern "C" void kernel_launch(void* const* d_in, const int* in_sizes, int n_in,
                              void* d_out, int out_size, void* d_ws, size_t ws_size,
                              hipStream_t stream) {
    (void)in_sizes; (void)n_in; (void)out_size; (void)ws_size;
    // setup_inputs order: x, lens, W, label_seqs, sos
    const float* x    = (const float*)d_in[0]; (void)x;    // provably unused
    const int*   lens = (const int*)  d_in[1]; (void)lens; // provably unused
    const float* W    = (const float*)d_in[2];
    const float* L    = (const float*)d_in[3];
    const float* sos  = (const float*)d_in[4];
    float* ws = (float*)d_ws;
    int*   out = (int*)d_out;

    sosw_kernel     <<<1,               256, 0, stream>>>(W, sos, ws);
    d0_kernel       <<<C_DIM,           256, 0, stream>>>(L, ws);
    top5_kernel     <<<1,                 1, 0, stream>>>(ws);
    beam_gemm_kernel<<<dim3(16, 2, NB),  32, 0, stream>>>(L, W, ws);
    rowsumsq_kernel <<<C_DIM,           256, 0, stream>>>(L, KFLAT, ws + OFF_SL);
    rowsumsq_kernel <<<NB,              256, 0, stream>>>(ws + OFF_P, KFLAT, ws + OFF_SP);
    score_gemm_kernel<<<63,              32, 0, stream>>>(L, ws);
    final_kernel    <<<1,               256, 0, stream>>>(ws, out);
}